// SparseEmbedding_42803644072658
// MI455X (gfx1250) — compile-verified
//
#include <hip/hip_runtime.h>

// ---------------------------------------------------------------------------
// SparseEmbedding forward (gather): out[r, :] = weight[idx[r], :]
//   weight : 1,000,000 x 128 fp32  (512 MB)
//   idx    : 819,200 int           (4096*200)
//   out    : 819,200 x 128 fp32    (419 MB)
// Memory-bound: ~839 MB moved -> ~36 us floor at 23.3 TB/s. Zero FLOPs, so
// no WMMA; the CDNA5 feature used is the async global<->LDS DMA path
// (GLOBAL_LOAD_ASYNC_TO_LDS_B128 / GLOBAL_STORE_ASYNC_FROM_LDS_B128,
// ordered with S_WAIT_ASYNCCNT). One row = 512 B = 32 lanes x b128, so a
// wave moves a whole row per async op and the payload never touches VGPRs.
// ---------------------------------------------------------------------------

#define EMB_D 128              // floats per row
#define ROWS_PER_WAVE 8        // 4 KB in flight per wave between waits
#define WAVES_PER_BLOCK 8      // 256 threads, wave32
#define ROWS_PER_BLOCK (ROWS_PER_WAVE * WAVES_PER_BLOCK)   // 64

typedef float v4f __attribute__((ext_vector_type(4)));
typedef int   v4i __attribute__((ext_vector_type(4)));

// Builtin pointer types per hipcc diagnostic:
//   param 0: int __attribute__((vector_size(16))) __device__*  (global, AS1)
//   param 1: LDS-side pointer (AS3), same pointee
typedef int v4i_b __attribute__((vector_size(16)));
typedef __attribute__((address_space(1))) v4i_b* gas_b128_t;
typedef __attribute__((address_space(3))) v4i_b* las_b128_t;

#if __has_builtin(__builtin_amdgcn_global_load_async_to_lds_b128) && \
    __has_builtin(__builtin_amdgcn_global_store_async_from_lds_b128) && \
    __has_builtin(__builtin_amdgcn_s_wait_asynccnt)
#define USE_ASYNC 1
#else
#define USE_ASYNC 0
#endif

#if USE_ASYNC
// Async DMA path: weight row -> LDS -> out, payload never touches VGPRs.
__global__ __launch_bounds__(256) void emb_gather_async(
    const float* __restrict__ w, const int* __restrict__ idx,
    float* __restrict__ out, int nrows) {
  __shared__ __align__(16) float lbuf[ROWS_PER_BLOCK * EMB_D];  // 32 KB/block
  const int lane = (int)threadIdx.x & 31;
  const int wid  = (int)threadIdx.x >> 5;
  const int base = ((int)blockIdx.x * WAVES_PER_BLOCK + wid) * ROWS_PER_WAVE;
  if (base >= nrows) return;

  float* lrow0 = &lbuf[wid * ROWS_PER_WAVE * EMB_D];
  const int laneOff = lane * 4;  // float offset of this lane's 16 B chunk

  if (base + ROWS_PER_WAVE <= nrows) {
    // ---- fast path: full wave-batch, no exec-mask churn ----
    // One vectorized fetch of all 8 row indices (2 x b128, 1 wait) instead
    // of 8 dependent load+wait round-trips.
    const v4i i0 = *(const v4i*)(idx + base);
    const v4i i1 = *(const v4i*)(idx + base + 4);
    int rs[ROWS_PER_WAVE] = {i0.x, i0.y, i0.z, i0.w, i1.x, i1.y, i1.z, i1.w};

#pragma unroll
    for (int k = 0; k < ROWS_PER_WAVE; ++k) {
      const float* gsrc = w + (size_t)(unsigned)rs[k] * EMB_D + laneOff;
      float* ldst = lrow0 + k * EMB_D + laneOff;
      __builtin_amdgcn_global_load_async_to_lds_b128(
          (gas_b128_t)gsrc, (las_b128_t)ldst, /*offset=*/0, /*cpol=*/0);
    }
    // LDS side of async ops is unordered -> drain before stores read LDS.
    __builtin_amdgcn_s_wait_asynccnt(0);
#pragma unroll
    for (int k = 0; k < ROWS_PER_WAVE; ++k) {
      float* gdst = out + (size_t)(base + k) * EMB_D + laneOff;
      float* lsrc = lrow0 + k * EMB_D + laneOff;
      __builtin_amdgcn_global_store_async_from_lds_b128(
          (gas_b128_t)gdst, (las_b128_t)lsrc, /*offset=*/0, /*cpol=*/0);
    }
    // S_ENDPGM's implicit wait-idle drains ASYNCcnt before wave retire.
  } else {
    // ---- tail path: per-row guards (rare) ----
    const int nvalid = nrows - base;
#pragma unroll
    for (int k = 0; k < ROWS_PER_WAVE; ++k) {
      if (k < nvalid) {
        const size_t r = (size_t)(unsigned)idx[base + k];
        const float* gsrc = w + r * EMB_D + laneOff;
        float* ldst = lrow0 + k * EMB_D + laneOff;
        __builtin_amdgcn_global_load_async_to_lds_b128(
            (gas_b128_t)gsrc, (las_b128_t)ldst, 0, 0);
      }
    }
    __builtin_amdgcn_s_wait_asynccnt(0);
#pragma unroll
    for (int k = 0; k < ROWS_PER_WAVE; ++k) {
      if (k < nvalid) {
        float* gdst = out + (size_t)(base + k) * EMB_D + laneOff;
        float* lsrc = lrow0 + k * EMB_D + laneOff;
        __builtin_amdgcn_global_store_async_from_lds_b128(
            (gas_b128_t)gdst, (las_b128_t)lsrc, 0, 0);
      }
    }
  }
}
#endif

// Fallback path (unused when async builtins exist): direct b128 gather
// through VGPRs with nontemporal stores.
__global__ __launch_bounds__(256) void emb_gather_direct(
    const float* __restrict__ w, const int* __restrict__ idx,
    float* __restrict__ out, int nrows) {
  const int lane = (int)threadIdx.x & 31;
  const int wid  = (int)threadIdx.x >> 5;
  const int base = ((int)blockIdx.x * WAVES_PER_BLOCK + wid) * ROWS_PER_WAVE;
  const v4f* w4 = (const v4f*)w;
  v4f* out4 = (v4f*)out;

#pragma unroll
  for (int k = 0; k < ROWS_PER_WAVE; ++k) {
    const int r = base + k;
    if (r < nrows) {
      const size_t row = (size_t)(unsigned)idx[r];
      v4f d = w4[row * (EMB_D / 4) + lane];
      __builtin_nontemporal_store(d, &out4[(size_t)r * (EMB_D / 4) + lane]);
    }
  }
}

extern "C" void kernel_launch(void* const* d_in, const int* in_sizes, int n_in,
                              void* d_out, int out_size, void* d_ws, size_t ws_size,
                              hipStream_t stream) {
  const float* w   = (const float*)d_in[0];   // weight [1e6 x 128] fp32
  const int*   idx = (const int*)d_in[1];     // indices, flat count in in_sizes[1]
  float*       out = (float*)d_out;           // [nrows x 128] fp32
  const int nrows  = in_sizes[1];
  const int blocks = (nrows + ROWS_PER_BLOCK - 1) / ROWS_PER_BLOCK;

#if USE_ASYNC
  emb_gather_async<<<blocks, 256, 0, stream>>>(w, idx, out, nrows);
#else
  emb_gather_direct<<<blocks, 256, 0, stream>>>(w, idx, out, nrows);
#endif
}